// AdditiveAttention_83717502534077
// MI455X (gfx1250) — compile-verified
//
#include <hip/hip_runtime.h>
#include <hip/hip_bf16.h>
#include <math.h>

// MI455X / gfx1250, wave32. Additive (Bahdanau) attention.
// B=4, Lq=Lk=256, d=512.

typedef __attribute__((ext_vector_type(2))) float v2f;
typedef __attribute__((ext_vector_type(4))) float v4f;
typedef __attribute__((ext_vector_type(8))) float v8f;

#define B_  4
#define LQ  256
#define LK  256
#define DM  512

__device__ __forceinline__ float fast_tanh(float x) {
#if __has_builtin(__builtin_amdgcn_tanhf)
  return __builtin_amdgcn_tanhf(x);
#else
  return tanhf(x);
#endif
}

// ---------------------------------------------------------------------------
// P[b,r,e] = sum_d X[b,r,d] * W[e,d] + bias[e]
// One wave per 16x16 output tile, V_WMMA_F32_16X16X4_F32 over d.
// 4 independent accumulators break the WMMA D->C dependency chain so the
// XDL pipe can overlap successive ops.
// A (16x4 f32): lane L holds A[L&15][2*(L>>4)+j], j=0,1  -> contiguous v2f.
// B (4x16 f32): lane L holds B[2*(L>>4)+j][L&15]; B[k][e] = W[e,k] -> v2f too.
// ---------------------------------------------------------------------------
__global__ void proj_wmma(const float* __restrict__ X, const float* __restrict__ W,
                          const float* __restrict__ bias, float* __restrict__ P) {
  const int lane  = threadIdx.x;        // 0..31
  const int rbase = blockIdx.x * 16;    // row tile
  const int ebase = blockIdx.y * 16;    // out-feature tile
  const int b     = blockIdx.z;
  const int half  = lane >> 4;          // 0 or 1
  const int l15   = lane & 15;
  const int kof   = half * 2;

  const float* Xrow = X + (size_t)(b * LQ + rbase + l15) * DM + kof;
  const float* Wrow = W + (size_t)(ebase + l15) * DM + kof;

  v8f acc0 = {}, acc1 = {}, acc2 = {}, acc3 = {};
  for (int k0 = 0; k0 < DM; k0 += 16) {
    v2f a0 = *reinterpret_cast<const v2f*>(Xrow + k0);
    v2f b0 = *reinterpret_cast<const v2f*>(Wrow + k0);
    v2f a1 = *reinterpret_cast<const v2f*>(Xrow + k0 + 4);
    v2f b1 = *reinterpret_cast<const v2f*>(Wrow + k0 + 4);
    v2f a2 = *reinterpret_cast<const v2f*>(Xrow + k0 + 8);
    v2f b2 = *reinterpret_cast<const v2f*>(Wrow + k0 + 8);
    v2f a3 = *reinterpret_cast<const v2f*>(Xrow + k0 + 12);
    v2f b3 = *reinterpret_cast<const v2f*>(Wrow + k0 + 12);
    acc0 = __builtin_amdgcn_wmma_f32_16x16x4_f32(false, a0, false, b0, (short)0, acc0, false, false);
    acc1 = __builtin_amdgcn_wmma_f32_16x16x4_f32(false, a1, false, b1, (short)0, acc1, false, false);
    acc2 = __builtin_amdgcn_wmma_f32_16x16x4_f32(false, a2, false, b2, (short)0, acc2, false, false);
    acc3 = __builtin_amdgcn_wmma_f32_16x16x4_f32(false, a3, false, b3, (short)0, acc3, false, false);
  }
  v8f acc = (acc0 + acc1) + (acc2 + acc3);

  const float bv = bias[ebase + l15];
  float* Prow = P + ((size_t)(b * LQ) + rbase) * DM + ebase + l15;
#pragma unroll
  for (int i = 0; i < 8; ++i) {
    int m = i + half * 8;               // C/D layout: VGPR i -> row i (+8 for upper lanes)
    Prow[(size_t)m * DM] = acc[i] + bv;
  }
}

// ---------------------------------------------------------------------------
// One block per (b,q): scores over all 256 k, softmax, write attn row.
// 8 waves; each wave owns one k at a time; 32 lanes split d. float4 loads
// (global b128 / ds b128) keep the loop TRANS-bound on v_tanh_f32.
// ---------------------------------------------------------------------------
__global__ void score_softmax(const float* __restrict__ Qp, const float* __restrict__ Kp,
                              const float* __restrict__ vw, const float* __restrict__ vb,
                              float* __restrict__ attn) {
  __shared__ float sQ[DM];
  __shared__ float sV[DM];
  __shared__ float sS[LK];
  __shared__ float sRedM[8];
  __shared__ float sRedS[8];

  const int t    = threadIdx.x;         // 0..255
  const int lane = t & 31;
  const int w    = t >> 5;              // wave 0..7
  const int b    = blockIdx.x / LQ;
  const int q    = blockIdx.x % LQ;

  const float* qrow = Qp + (size_t)(b * LQ + q) * DM;
  sQ[t]       = qrow[t];
  sQ[t + 256] = qrow[t + 256];
  sV[t]       = vw[t];
  sV[t + 256] = vw[t + 256];
  __syncthreads();

  const v4f* sQ4 = reinterpret_cast<const v4f*>(sQ);
  const v4f* sV4 = reinterpret_cast<const v4f*>(sV);

  const float vbias = vb[0];
  for (int k0 = 0; k0 < LK; k0 += 8) {
    const int k = k0 + w;
    const v4f* krow4 = reinterpret_cast<const v4f*>(Kp + (size_t)(b * LK + k) * DM);
    float acc = 0.0f;
#pragma unroll
    for (int i = 0; i < DM / 128; ++i) {        // 4 iterations of float4
      const int d4 = lane + i * 32;
      v4f kk = krow4[d4];
      v4f qq = sQ4[d4];
      v4f vv = sV4[d4];
      acc += vv[0] * fast_tanh(qq[0] + kk[0]);
      acc += vv[1] * fast_tanh(qq[1] + kk[1]);
      acc += vv[2] * fast_tanh(qq[2] + kk[2]);
      acc += vv[3] * fast_tanh(qq[3] + kk[3]);
    }
#pragma unroll
    for (int o = 16; o > 0; o >>= 1) acc += __shfl_down(acc, o, 32);
    if (lane == 0) sS[k] = acc + vbias;
  }
  __syncthreads();

  // softmax over the 256 scores of this row
  const float x = sS[t];
  float m = x;
#pragma unroll
  for (int o = 16; o > 0; o >>= 1) m = fmaxf(m, __shfl_down(m, o, 32));
  if (lane == 0) sRedM[w] = m;
  __syncthreads();
  if (t < 8) {
    float mm = sRedM[t];
#pragma unroll
    for (int o = 4; o > 0; o >>= 1) mm = fmaxf(mm, __shfl_down(mm, o, 8));
    if (t == 0) sRedM[0] = mm;
  }
  __syncthreads();
  m = sRedM[0];

  const float e = __expf(x - m);
  float s = e;
#pragma unroll
  for (int o = 16; o > 0; o >>= 1) s += __shfl_down(s, o, 32);
  if (lane == 0) sRedS[w] = s;
  __syncthreads();
  if (t < 8) {
    float ss = sRedS[t];
#pragma unroll
    for (int o = 4; o > 0; o >>= 1) ss += __shfl_down(ss, o, 8);
    if (t == 0) sRedS[0] = ss;
  }
  __syncthreads();

  attn[(size_t)(b * LQ + q) * LK + t] = e / sRedS[0];
}

// ---------------------------------------------------------------------------
// out[b,q,e] = sum_k attn[b,q,k] * value[b,k,e]   (WMMA f32 16x16x4 over k,
// 4 independent accumulator chains)
// ---------------------------------------------------------------------------
__global__ void out_wmma(const float* __restrict__ attn, const float* __restrict__ V,
                         float* __restrict__ out) {
  const int lane  = threadIdx.x;
  const int qbase = blockIdx.x * 16;
  const int ebase = blockIdx.y * 16;
  const int b     = blockIdx.z;
  const int half  = lane >> 4;
  const int l15   = lane & 15;
  const int kof   = half * 2;

  const float* arow = attn + (size_t)(b * LQ + qbase + l15) * LK + kof;
  const float* vcol = V + (size_t)(b * LK) * DM + ebase + l15;

  v8f acc0 = {}, acc1 = {}, acc2 = {}, acc3 = {};
  for (int k0 = 0; k0 < LK; k0 += 16) {
#pragma unroll
    for (int j = 0; j < 4; ++j) {
      v2f a = *reinterpret_cast<const v2f*>(arow + k0 + 4 * j);
      v2f bm;
      bm[0] = vcol[(size_t)(k0 + 4 * j + kof) * DM];
      bm[1] = vcol[(size_t)(k0 + 4 * j + kof + 1) * DM];
      switch (j) {
        case 0: acc0 = __builtin_amdgcn_wmma_f32_16x16x4_f32(false, a, false, bm, (short)0, acc0, false, false); break;
        case 1: acc1 = __builtin_amdgcn_wmma_f32_16x16x4_f32(false, a, false, bm, (short)0, acc1, false, false); break;
        case 2: acc2 = __builtin_amdgcn_wmma_f32_16x16x4_f32(false, a, false, bm, (short)0, acc2, false, false); break;
        case 3: acc3 = __builtin_amdgcn_wmma_f32_16x16x4_f32(false, a, false, bm, (short)0, acc3, false, false); break;
      }
    }
  }
  v8f acc = (acc0 + acc1) + (acc2 + acc3);

  float* orow = out + ((size_t)(b * LQ) + qbase) * DM + ebase + l15;
#pragma unroll
  for (int i = 0; i < 8; ++i) {
    int m = i + half * 8;
    orow[(size_t)m * DM] = acc[i];
  }
}

// ---------------------------------------------------------------------------
extern "C" void kernel_launch(void* const* d_in, const int* in_sizes, int n_in,
                              void* d_out, int out_size, void* d_ws, size_t ws_size,
                              hipStream_t stream) {
  const float* query = (const float*)d_in[0];
  const float* key   = (const float*)d_in[1];
  const float* value = (const float*)d_in[2];
  const float* Wq    = (const float*)d_in[3];
  const float* bq    = (const float*)d_in[4];
  const float* Wk    = (const float*)d_in[5];
  const float* bk    = (const float*)d_in[6];
  const float* vw    = (const float*)d_in[7];
  const float* vb    = (const float*)d_in[8];

  float* out  = (float*)d_out;                 // [B, LQ, DM]
  float* attn = out + (size_t)B_ * LQ * DM;    // [B, LQ, LK]

  float* Qp = (float*)d_ws;                    // [B, LQ, DM]
  float* Kp = Qp + (size_t)B_ * LQ * DM;       // [B, LK, DM]

  dim3 gp(LQ / 16, DM / 16, B_);
  proj_wmma<<<gp, 32, 0, stream>>>(query, Wq, bq, Qp);
  proj_wmma<<<gp, 32, 0, stream>>>(key,   Wk, bk, Kp);

  score_softmax<<<dim3(B_ * LQ), 256, 0, stream>>>(Qp, Kp, vw, vb, attn);

  out_wmma<<<dim3(LQ / 16, DM / 16, B_), 32, 0, stream>>>(attn, value, out);
}